// SingleModalFactorization_77567109366474
// MI455X (gfx1250) — compile-verified
//
#include <hip/hip_runtime.h>
#include <stdint.h>

// Problem constants from the reference.
constexpr int S_DIM = 16384;
constexpr int N_DIM = 2048;
constexpr int C_DIM = 8;

constexpr int PAD   = 64;                 // zero guard on each side of a component row
constexpr int ROWP  = N_DIM + 2 * PAD;    // 2176 floats per padded row
constexpr int RPB   = 16;                 // samples (rows of out) per block
constexpr int TPB   = 256;                // 8 wave32 waves

#define GLOBAL_AS __attribute__((address_space(1)))
#define LDS_AS    __attribute__((address_space(3)))

typedef int   v4i __attribute__((ext_vector_type(4)));
typedef float v2f __attribute__((ext_vector_type(2)));

#if defined(__gfx1250__) && __has_builtin(__builtin_amdgcn_global_load_async_to_lds_b128)
#define SMF_USE_ASYNC_LDS 1
#else
#define SMF_USE_ASYNC_LDS 0
#endif

__global__ __launch_bounds__(TPB) void SingleModalFactorization_kernel(
    const float* __restrict__ components,     // [C][N]
    const float* __restrict__ contributions,  // [S][C]
    const float* __restrict__ shift,          // [S][C]
    float* __restrict__ out)                  // [S][N]
{
    __shared__ __align__(16) float comp[C_DIM * ROWP];  // zero-padded component rows
    __shared__ float a0s[RPB][C_DIM];
    __shared__ float a1s[RPB][C_DIM];
    __shared__ int   fss[RPB][C_DIM];

    const int tid = threadIdx.x;
    const int s0  = blockIdx.x * RPB;

    // ---- 1. kick off the async global->LDS DMA of the component rows FIRST,
    //         so the ASYNCcnt-tracked transfers overlap the setup work below.
#if SMF_USE_ASYNC_LDS
    {
        // 4096 x b128 transfers total, 16 per thread.
        #pragma unroll
        for (int k = 0; k < (C_DIM * N_DIM) / (4 * TPB); ++k) {
            const int e   = (tid + k * TPB) * 4;       // flat float index, multiple of 4
            const int c   = e / N_DIM;
            const int col = e % N_DIM;
            // addrspace(1) int4* built via inttoptr (global flat addr == AS1 addr)
            GLOBAL_AS v4i* gp = (GLOBAL_AS v4i*)(uintptr_t)(components + e);
            // low 32 bits of the flat LDS address == LDS byte offset (AS3 is 32-bit)
            const uint32_t lds_off =
                (uint32_t)(uintptr_t)(comp + c * ROWP + PAD + col);
            LDS_AS v4i* lp = (LDS_AS v4i*)(uintptr_t)lds_off;
            __builtin_amdgcn_global_load_async_to_lds_b128(gp, lp, 0, 0);
        }
    }
#else
    for (int k = tid; k < (C_DIM * N_DIM) / 4; k += TPB) {
        const float4 v = ((const float4*)components)[k];
        const int e   = k * 4;
        const int c   = e / N_DIM;
        const int col = e % N_DIM;
        *(float4*)&comp[c * ROWP + PAD + col] = v;
    }
#endif

    // ---- 2. zero the guard zones: 1024 floats = 256 float4 -> ONE b128 store/thread.
    //         Guard region is disjoint from the async-DMA destination region.
    {
        const int c = tid >> 5;          // component row (8 rows x 32 float4 chunks)
        const int j = tid & 31;          // chunk within row: 16 front + 16 back
        const int idx4 = (j < 16) ? j : (512 + j);   // front: 0..15, back: 528..543
        float4* comp4 = (float4*)comp;
        comp4[c * (ROWP / 4) + idx4] = make_float4(0.f, 0.f, 0.f, 0.f);
    }

    // ---- 3. per-(sample, component) interpolation constants ----
    // floor(n + shift) == n + floor(shift) for integer n, so the lerp weights
    // and the integer offset are per-(s,c) scalars.
    if (tid < RPB * C_DIM) {
        const int r = tid / C_DIM;
        const int c = tid % C_DIM;
        const int s = s0 + r;
        const float sh = shift[s * C_DIM + c];
        const float ct = contributions[s * C_DIM + c];
        const float ff = floorf(sh);
        const float w1 = sh - ff;        // frac in [0,1)
        const float a1 = ct * w1;
        a1s[r][c] = a1;
        a0s[r][c] = ct - a1;             // ct * (1 - w1)
        int fi = (int)ff;
        fi = fi < -(N_DIM + 2 * PAD) ? -(N_DIM + 2 * PAD) : fi;
        fi = fi >  (N_DIM + 2 * PAD) ?  (N_DIM + 2 * PAD) : fi;
        fss[r][c] = fi;
    }

#if SMF_USE_ASYNC_LDS
#if __has_builtin(__builtin_amdgcn_s_wait_asynccnt)
    __builtin_amdgcn_s_wait_asynccnt(0);
#else
    asm volatile("s_wait_asynccnt 0" ::: "memory");
#endif
#endif
    __syncthreads();

    // ---- 4. main loop: 2 adjacent channels per thread, 3 gathers -> 4 FMAs per c ----
    constexpr int IDX_LO = -PAD;
    constexpr int IDX_HI = N_DIM + PAD - 3;  // need taps idx, idx+1, idx+2 in bounds

    for (int r = 0; r < RPB; ++r) {
        float a0[C_DIM], a1[C_DIM];
        int   fs[C_DIM];
        #pragma unroll
        for (int c = 0; c < C_DIM; ++c) {    // broadcast LDS reads (no conflicts)
            a0[c] = a0s[r][c];
            a1[c] = a1s[r][c];
            fs[c] = fss[r][c];
        }
        const int s = s0 + r;
        float* orow = out + (size_t)s * N_DIM;

        #pragma unroll
        for (int p = 0; p < N_DIM / (2 * TPB); ++p) {    // 4 passes
            const int n0 = 2 * tid + p * 2 * TPB;
            float acc0 = 0.0f, acc1 = 0.0f;
            #pragma unroll
            for (int c = 0; c < C_DIM; ++c) {
                int idx = n0 + fs[c];
                idx = idx < IDX_LO ? IDX_LO : idx;       // clamp -> lands in zero guard
                idx = idx > IDX_HI ? IDX_HI : idx;       // when truly out of range
                const float* row = &comp[c * ROWP + PAD];
                // lane stride = 2 floats -> 32 lanes hit 32 distinct banks: conflict-free
                const float g0 = row[idx];
                const float g1 = row[idx + 1];
                const float g2 = row[idx + 2];
                acc0 = fmaf(a0[c], g0, acc0);
                acc0 = fmaf(a1[c], g1, acc0);
                acc1 = fmaf(a0[c], g1, acc1);
                acc1 = fmaf(a1[c], g2, acc1);
            }
            // streaming output: non-temporal store, coalesced b64 per lane
            v2f val = {acc0, acc1};
            __builtin_nontemporal_store(val, (v2f*)(orow + n0));
        }
    }
}

extern "C" void kernel_launch(void* const* d_in, const int* in_sizes, int n_in,
                              void* d_out, int out_size, void* d_ws, size_t ws_size,
                              hipStream_t stream) {
    // d_in[0] = inputs (S*N)      -- UNUSED by the reference computation
    // d_in[1] = components (C*N)
    // d_in[2] = contributions (S*C)
    // d_in[3] = shift (S*C)
    const float* components    = (const float*)d_in[1];
    const float* contributions = (const float*)d_in[2];
    const float* shiftp        = (const float*)d_in[3];
    float* outp                = (float*)d_out;

    const dim3 grid(S_DIM / RPB);   // 1024 blocks
    const dim3 block(TPB);          // 256 threads = 8 wave32
    SingleModalFactorization_kernel<<<grid, block, 0, stream>>>(
        components, contributions, shiftp, outp);
}